// GritMessagePassing_44805098832270
// MI455X (gfx1250) — compile-verified
//
#include <hip/hip_runtime.h>
#include <math.h>

typedef __attribute__((ext_vector_type(2))) float v2f;
typedef __attribute__((ext_vector_type(8))) float v8f;

#define N_NODES 50000
#define N_EDGES 800000
#define HIDDEN  64
#define HEADS   8
#define ADIM    8
#define HD      64      // HEADS*ADIM
#define CLAMP_V 5.0f

// ---------------------------------------------------------------------------
// Generic zero-fill (grid-stride)
// ---------------------------------------------------------------------------
__global__ void zero_f32(float* __restrict__ p, long long n) {
    long long i = (long long)blockIdx.x * blockDim.x + threadIdx.x;
    long long stride = (long long)gridDim.x * blockDim.x;
    for (; i < n; i += stride) p[i] = 0.0f;
}

// ---------------------------------------------------------------------------
// Y[M,Nout] = X[M,64] @ W[Nout,64]^T + bias[Nout]
//
// One wave32 per 16-row strip of M. The 16 A-fragments are loaded ONCE into
// registers and reused across all Nout/16 column tiles (X streamed exactly
// once from HBM). Per tile, ALL 16 B-fragments are loaded into a register
// array first (one load clause + one wait), then the 16 WMMAs run
// back-to-back -- avoids the per-WMMA load-latency serialization.
// Bias is folded into the accumulator init (bias depends only on N=lane%16,
// identical across all 8 C VGPRs).
//
// f32 WMMA fragment layouts (ISA 7.12.2):
//   A 16x4:  lanes 0-15 -> M=lane,    VGPR0=K0 VGPR1=K1
//            lanes16-31 -> M=lane-16, VGPR0=K2 VGPR1=K3
//   B 4x16:  mirrored (lane selects N, half selects K pair)
//   C/D 16x16: VGPR v -> M=v (lanes 0-15) / M=v+8 (lanes 16-31), N=lane%16
// ---------------------------------------------------------------------------
template <int NOUT>
__global__ void gemm_xwT_wmma(const float* __restrict__ X,
                              const float* __restrict__ W,
                              const float* __restrict__ bias,
                              float* __restrict__ Y,
                              int tilesM) {
    int tm = blockIdx.x * blockDim.y + threadIdx.y;   // one wave per M strip
    if (tm >= tilesM) return;                          // wave-uniform branch

    int lane = threadIdx.x;          // 0..31
    int half = lane >> 4;            // selects K pair {0,1} vs {2,3}
    int idx  = lane & 15;

    // Load all 16 A-fragments for this 16-row strip (held in registers).
    const float* arow = X + (long long)(tm * 16 + idx) * HIDDEN + half * 2;
    v2f afrag[16];
#pragma unroll
    for (int i = 0; i < 16; ++i)
        afrag[i] = *(const v2f*)(arow + 4 * i);

    constexpr int tilesN = NOUT >> 4;
    int orow0 = tm * 16 + half * 8;

#pragma unroll
    for (int tn = 0; tn < tilesN; ++tn) {             // wave-uniform loop
        const float* brow = W + (long long)(tn * 16 + idx) * HIDDEN + half * 2;

        // batch all B loads for this tile -> one clause, one wait
        v2f bfrag[16];
#pragma unroll
        for (int i = 0; i < 16; ++i)
            bfrag[i] = *(const v2f*)(brow + 4 * i);

        int ocol = tn * 16 + idx;
        float bv = bias[ocol];
        v8f acc = {bv, bv, bv, bv, bv, bv, bv, bv};   // bias folded into C

#pragma unroll
        for (int i = 0; i < 16; ++i) {
            acc = __builtin_amdgcn_wmma_f32_16x16x4_f32(
                /*neg_a=*/false, afrag[i], /*neg_b=*/false, bfrag[i],
                /*c_mod=*/(short)0, acc, /*reuse_a=*/false, /*reuse_b=*/false);
        }

#pragma unroll
        for (int v = 0; v < 8; ++v) {
            Y[(long long)(orow0 + v) * NOUT + ocol] = acc[v];
        }
    }
}

// ---------------------------------------------------------------------------
// order-preserving float<->uint encoding for atomicMax on mixed-sign floats
// ---------------------------------------------------------------------------
__device__ __forceinline__ unsigned enc_f32(float x) {
    unsigned b = __float_as_uint(x);
    return (b & 0x80000000u) ? ~b : (b | 0x80000000u);
}
__device__ __forceinline__ float dec_f32(unsigned k) {
    unsigned b = (k & 0x80000000u) ? (k ^ 0x80000000u) : ~k;
    return __uint_as_float(b);
}

// ---------------------------------------------------------------------------
// Pass 1: per (edge, head): conn, clamped score, atomicMax of score per (dst,h)
// ---------------------------------------------------------------------------
__global__ void edge_pass1(const float* __restrict__ qkv,   // [N,192]
                           const float* __restrict__ Eh,    // [E,128]
                           const int*   __restrict__ ei,    // [2,E]
                           const float* __restrict__ Aw,    // [ADIM,HEADS]
                           float*       __restrict__ connOut, // [E,64]
                           float*       __restrict__ scoreW,  // [E,8]
                           unsigned*    __restrict__ mEnc)    // [N,8]
{
    int t = blockIdx.x * blockDim.x + threadIdx.x;
    if (t >= N_EDGES * HEADS) return;
    int e = t >> 3, h = t & 7;
    int dst = ei[e], src = ei[N_EDGES + e];

    const float* q  = qkv + (long long)dst * 192 + h * ADIM;        // Qh
    const float* kk = qkv + (long long)src * 192 + 64 + h * ADIM;   // Kh
    const float* ew = Eh  + (long long)e * 128 + h * ADIM;          // Ew
    const float* eb = Eh  + (long long)e * 128 + 64 + h * ADIM;     // Eb

    float score = 0.0f;
#pragma unroll
    for (int d = 0; d < ADIM; ++d) {
        float c1 = (q[d] + kk[d]) * ew[d];
        float c2 = copysignf(sqrtf(fabsf(c1)), c1);   // sqrt_relu(x)-sqrt_relu(-x)
        float c  = fmaxf(c2 + eb[d], 0.0f);
        connOut[(long long)e * HD + h * ADIM + d] = c;
        score += c * Aw[d * HEADS + h];
    }
    score = fminf(fmaxf(score, -CLAMP_V), CLAMP_V);
    scoreW[(long long)e * HEADS + h] = score;
    atomicMax(&mEnc[dst * HEADS + h], enc_f32(score));
}

// ---------------------------------------------------------------------------
// Pass 2: denominator  s[dst,h] += exp(score - m[dst,h])
// ---------------------------------------------------------------------------
__global__ void edge_pass2(const float*    __restrict__ scoreW,
                           const unsigned* __restrict__ mEnc,
                           const int*      __restrict__ ei,
                           float*          __restrict__ s)
{
    int t = blockIdx.x * blockDim.x + threadIdx.x;
    if (t >= N_EDGES * HEADS) return;
    int e = t >> 3, h = t & 7;
    int dst = ei[e];
    float m = dec_f32(mEnc[dst * HEADS + h]);
    float ev = expf(scoreW[(long long)e * HEADS + h] - m);
    atomicAdd(&s[dst * HEADS + h], ev);
}

// ---------------------------------------------------------------------------
// Pass 3: alpha = exp(score-m)/(s+eps); scatter V*alpha -> No, conn*alpha -> rowV
// ---------------------------------------------------------------------------
__global__ void edge_pass3(const float*    __restrict__ qkv,
                           const float*    __restrict__ connOut,
                           const float*    __restrict__ scoreW,
                           const unsigned* __restrict__ mEnc,
                           const float*    __restrict__ s,
                           const int*      __restrict__ ei,
                           float*          __restrict__ No,      // [N,64] (agg)
                           float*          __restrict__ rowVacc) // [N,64]
{
    int t = blockIdx.x * blockDim.x + threadIdx.x;
    if (t >= N_EDGES * HEADS) return;
    int e = t >> 3, h = t & 7;
    int dst = ei[e], src = ei[N_EDGES + e];

    float m = dec_f32(mEnc[dst * HEADS + h]);
    float alpha = expf(scoreW[(long long)e * HEADS + h] - m) /
                  (s[dst * HEADS + h] + 1e-16f);

    const float* v = qkv + (long long)src * 192 + 128 + h * ADIM;   // Vh
    const float* c = connOut + (long long)e * HD + h * ADIM;
#pragma unroll
    for (int d = 0; d < ADIM; ++d) {
        atomicAdd(&No[(long long)dst * HD + h * ADIM + d], v[d] * alpha);
        atomicAdd(&rowVacc[(long long)dst * HD + h * ADIM + d], c[d] * alpha);
    }
}

// ---------------------------------------------------------------------------
// Pass 4: No[n,h,:] += rowVacc[n,h,:] @ Bw[:,h,:]   (8x8 per head)
// ---------------------------------------------------------------------------
__global__ void node_finalize(const float* __restrict__ rowVacc, // [N,64]
                              const float* __restrict__ Bw,      // [ADIM,HEADS,ADIM]
                              float*       __restrict__ No)      // [N,64]
{
    int t = blockIdx.x * blockDim.x + threadIdx.x;
    if (t >= N_NODES * HEADS) return;
    int n = t >> 3, h = t & 7;

    float rv[ADIM];
#pragma unroll
    for (int d = 0; d < ADIM; ++d)
        rv[d] = rowVacc[(long long)n * HD + h * ADIM + d];

#pragma unroll
    for (int cidx = 0; cidx < ADIM; ++cidx) {
        float acc = 0.0f;
#pragma unroll
        for (int d = 0; d < ADIM; ++d)
            acc += rv[d] * Bw[(d * HEADS + h) * ADIM + cidx];
        No[(long long)n * HD + h * ADIM + cidx] += acc;
    }
}

// ---------------------------------------------------------------------------
extern "C" void kernel_launch(void* const* d_in, const int* in_sizes, int n_in,
                              void* d_out, int out_size, void* d_ws, size_t ws_size,
                              hipStream_t stream) {
    const float* x          = (const float*)d_in[0];  // [N,64]
    const float* conn_feat  = (const float*)d_in[1];  // [E,64]
    const int*   edge_index = (const int*)  d_in[2];  // [2,E]
    const float* qkv_weight = (const float*)d_in[3];  // [192,64]
    const float* qkv_bias   = (const float*)d_in[4];  // [192]
    const float* E_weight   = (const float*)d_in[5];  // [128,64]
    const float* E_bias     = (const float*)d_in[6];  // [128]
    const float* Aw         = (const float*)d_in[7];  // [8,8,1]
    const float* Bw         = (const float*)d_in[8];  // [8,8,8]

    float* out = (float*)d_out;
    float* No      = out;                            // [N,64]
    float* connOut = out + (long long)N_NODES * HD;  // [E,64]

    // workspace layout (floats)
    float* ws = (float*)d_ws;
    float* qkv     = ws;                                   // N*192
    float* Eh      = qkv + (long long)N_NODES * 192;       // E*128
    float* scoreW  = Eh + (long long)N_EDGES * 128;        // E*8
    unsigned* mEnc = (unsigned*)(scoreW + (long long)N_EDGES * HEADS); // N*8
    float* s       = (float*)mEnc + (long long)N_NODES * HEADS;        // N*8
    float* rowVacc = s + (long long)N_NODES * HEADS;       // N*64

    // 0) zero accumulators: No, and contiguous [mEnc | s | rowVacc]
    zero_f32<<<2048, 256, 0, stream>>>(No, (long long)N_NODES * HD);
    zero_f32<<<2048, 256, 0, stream>>>((float*)mEnc,
        (long long)N_NODES * (HEADS + HEADS + HD));

    // 1) qkv GEMM: [50000,64] x [64,192] -> 3125 M-strips, 12 N tiles each
    {
        int tilesM = N_NODES / 16;
        dim3 blk(32, 8);   // 8 waves / block
        gemm_xwT_wmma<192><<<(tilesM + 7) / 8, blk, 0, stream>>>(
            x, qkv_weight, qkv_bias, qkv, tilesM);
    }
    // 2) Eh GEMM: [800000,64] x [64,128] -> 50000 M-strips, 8 N tiles each
    {
        int tilesM = N_EDGES / 16;
        dim3 blk(32, 8);
        gemm_xwT_wmma<128><<<(tilesM + 7) / 8, blk, 0, stream>>>(
            conn_feat, E_weight, E_bias, Eh, tilesM);
    }

    const int ET = N_EDGES * HEADS;           // 6.4M threads
    edge_pass1<<<(ET + 255) / 256, 256, 0, stream>>>(
        qkv, Eh, edge_index, Aw, connOut, scoreW, mEnc);
    edge_pass2<<<(ET + 255) / 256, 256, 0, stream>>>(
        scoreW, mEnc, edge_index, s);
    edge_pass3<<<(ET + 255) / 256, 256, 0, stream>>>(
        qkv, connOut, scoreW, mEnc, s, edge_index, No, rowVacc);

    const int NT = N_NODES * HEADS;
    node_finalize<<<(NT + 255) / 256, 256, 0, stream>>>(rowVacc, Bw, No);
}